// Net_28913719836943
// MI455X (gfx1250) — compile-verified
//
#include <hip/hip_runtime.h>
#include <hip/hip_bf16.h>

typedef float v2f __attribute__((ext_vector_type(2)));
typedef float v8f __attribute__((ext_vector_type(8)));

#define KCHUNK 112   // K-rows of B staged in LDS per chunk (784 = 7*112; 64 <= 112; %16==0)

// ---------------------------------------------------------------- utilities
__global__ void k_zero(float* __restrict__ p, size_t n) {
    size_t i = (size_t)blockIdx.x * blockDim.x + threadIdx.x;
    size_t stride = (size_t)gridDim.x * blockDim.x;
    for (; i < n; i += stride) p[i] = 0.0f;
}

__global__ void k_deg(float* __restrict__ deg, const int* __restrict__ col,
                      const float* __restrict__ w, int E) {
    int e = blockIdx.x * blockDim.x + threadIdx.x;
    if (e < E) atomicAdd(&deg[col[e]], w[e]);
}

__global__ void k_dinv(float* __restrict__ d, int N) {
    int i = blockIdx.x * blockDim.x + threadIdx.x;
    if (i < N) d[i] = rsqrtf(d[i] + 1.0f);   // deg >= 1 (self loop)
}

__global__ void k_norm(float* __restrict__ norm, const float* __restrict__ dinv,
                       const int* __restrict__ row, const int* __restrict__ col,
                       const float* __restrict__ w, int E) {
    int e = blockIdx.x * blockDim.x + threadIdx.x;
    if (e < E) norm[e] = dinv[row[e]] * w[e] * dinv[col[e]];
}

// ------------------------------------------------- WMMA f32 GEMM: D = A @ B
// A: [M x lda] (cols 0..Kw-1 valid; rows clamped for OOB lanes — their D rows
//    are never stored, and D row m depends only on A row m, so clamping is safe)
// B: [Kw x Nw] global, staged zero-padded into LDS as [Kpad x 64]
// D: [M x ldd], block computes rows_per_block = 16*(8/n_col_tiles) rows
// LDS layout pair-interleaved: elem(k,n) at ((k>>1)*64 + n)*2 + (k&1)
__global__ void k_gemm_wmma(const float* __restrict__ A, int lda, int M,
                            const float* __restrict__ B, int Kw, int Nw, int Kpad,
                            float* __restrict__ D, int ldd, int n_col_tiles) {
    extern __shared__ float ldsB[];
    const int lane  = threadIdx.x & 31;
    const int wave  = threadIdx.x >> 5;
    const int rows_per_block = 16 * (8 / n_col_tiles);
    const int mtile = wave / n_col_tiles;
    const int ntile = wave % n_col_tiles;
    const int mbase = blockIdx.x * rows_per_block + mtile * 16;
    const int nbase = ntile * 16;
    const int l15   = lane & 15;
    const int khalf = (lane >> 4) << 1;          // 0 for lanes 0-15, 2 for 16-31
    int m           = mbase + l15;               // A row for this lane
    if (m >= M) m = M - 1;                       // clamp: unconditional loads
    const int n     = nbase + l15;               // B/D col for this lane
    const float* __restrict__ Arow = A + (size_t)m * lda;

    v8f acc0 = {}, acc1 = {}, acc2 = {}, acc3 = {};

    for (int k0 = 0; k0 < Kpad; k0 += KCHUNK) {
        const int rows = (Kpad - k0) < KCHUNK ? (Kpad - k0) : KCHUNK; // multiple of 16
        // cooperative zero-padded stage of B chunk into LDS
        for (int idx = threadIdx.x; idx < rows * 64; idx += blockDim.x) {
            int kr = idx >> 6, nn = idx & 63;
            int kg = k0 + kr;
            float v = (kg < Kw && nn < Nw) ? B[(size_t)kg * Nw + nn] : 0.0f;
            ldsB[(((kr >> 1) * 64 + nn) << 1) + (kr & 1)] = v;
        }
        __syncthreads();

        const float* ap = Arow + k0 + khalf;
        for (int kk = 0; kk < rows; kk += 16) {
            const int ka = kk + khalf;           // even
            v2f a0 = *(const v2f*)(ap + kk);
            v2f a1 = *(const v2f*)(ap + kk + 4);
            v2f a2 = *(const v2f*)(ap + kk + 8);
            v2f a3 = *(const v2f*)(ap + kk + 12);
            v2f b0 = *(const v2f*)&ldsB[(((ka >> 1)      * 64 + n) << 1)];
            v2f b1 = *(const v2f*)&ldsB[((((ka >> 1) + 2) * 64 + n) << 1)];
            v2f b2 = *(const v2f*)&ldsB[((((ka >> 1) + 4) * 64 + n) << 1)];
            v2f b3 = *(const v2f*)&ldsB[((((ka >> 1) + 6) * 64 + n) << 1)];
            acc0 = __builtin_amdgcn_wmma_f32_16x16x4_f32(
                       false, a0, false, b0, (short)0, acc0, false, false);
            acc1 = __builtin_amdgcn_wmma_f32_16x16x4_f32(
                       false, a1, false, b1, (short)0, acc1, false, false);
            acc2 = __builtin_amdgcn_wmma_f32_16x16x4_f32(
                       false, a2, false, b2, (short)0, acc2, false, false);
            acc3 = __builtin_amdgcn_wmma_f32_16x16x4_f32(
                       false, a3, false, b3, (short)0, acc3, false, false);
        }
        __syncthreads();
    }

    v8f acc = (acc0 + acc1) + (acc2 + acc3);
    const int row0 = mbase + 8 * (lane >> 4);    // D: VGPR r -> M = r + 8*(lane>=16)
    float* dp = D + (size_t)row0 * ldd + n;      // single base-address computation
    if (row0 + 7 < M) {                          // fast path: whole column in range
#pragma unroll
        for (int r = 0; r < 8; ++r) { *dp = acc[r]; dp += ldd; }
    } else {
#pragma unroll
        for (int r = 0; r < 8; ++r) { if (row0 + r < M) *dp = acc[r]; dp += ldd; }
    }
}

// ------------------------------- edge scatter: agg[col] += norm * h[row]
// stride/2 threads per edge; thread f handles features f and f+stride/2.
// Requires stride/2 <= fused (64/2=32<=50, 16/2=8<=10).
__global__ void k_scatter(float* __restrict__ agg, const float* __restrict__ h,
                          const float* __restrict__ norm,
                          const int* __restrict__ row, const int* __restrict__ col,
                          int E, int fshift, int fused) {
    const int tshift = fshift - 1;               // log2(threads per edge)
    long long t = (long long)blockIdx.x * blockDim.x + threadIdx.x;
    long long e = t >> tshift;
    int f = (int)(t & ((1 << tshift) - 1));
    if (e >= E) return;
    const int stride = 1 << fshift;
    int r = row[e], c = col[e];
    float w = norm[e];
    const float* hb = h + (size_t)r * stride;
    float*       ab = agg + (size_t)c * stride;
    atomicAdd(&ab[f], w * hb[f]);                // f < stride/2 <= fused always
    int f2 = f + (stride >> 1);
    if (f2 < fused) atomicAdd(&ab[f2], w * hb[f2]);
}

// ---------------------------- combine: agg = [relu](agg + dinv^2 * h + bias), pad -> 0
__global__ void k_combine(float* __restrict__ agg, const float* __restrict__ h,
                          const float* __restrict__ dinv, const float* __restrict__ bias,
                          int N, int fshift, int nout, int do_relu) {
    long long t = (long long)blockIdx.x * blockDim.x + threadIdx.x;
    const int stride = 1 << fshift;
    long long i = t >> fshift;
    int f = (int)(t & (stride - 1));
    if (i >= N) return;
    float d = dinv[i];
    size_t idx = (size_t)i * stride + f;
    float v = agg[idx] + d * d * h[idx] + ((f < nout) ? bias[f] : 0.0f);
    if (f >= nout) v = 0.0f;                 // keep K-padding exactly zero
    if (do_relu) v = fmaxf(v, 0.0f);
    agg[idx] = v;
}

// ------------------------------------------------------ per-node log_softmax (10 cls)
__global__ void k_logsoftmax(const float* __restrict__ z, float* __restrict__ out, int N) {
    int i = blockIdx.x * blockDim.x + threadIdx.x;
    if (i >= N) return;
    float v[10];
    float mx = -3.402823466e38f;
#pragma unroll
    for (int c = 0; c < 10; ++c) { v[c] = z[(size_t)i * 16 + c]; mx = fmaxf(mx, v[c]); }
    float s = 0.0f;
#pragma unroll
    for (int c = 0; c < 10; ++c) s += __expf(v[c] - mx);
    float l = __logf(s) + mx;
#pragma unroll
    for (int c = 0; c < 10; ++c) out[(size_t)i * 10 + c] = v[c] - l;
}

// ---------------------------------------------------------------------------- driver
extern "C" void kernel_launch(void* const* d_in, const int* in_sizes, int n_in,
                              void* d_out, int out_size, void* d_ws, size_t ws_size,
                              hipStream_t stream) {
    const int FEAT = 784, HID = 50, CLS = 10, NL = 4;
    const int N = in_sizes[0] / FEAT;
    const int E = in_sizes[2];

    const float* x   = (const float*)d_in[0];
    const int*   ei  = (const int*)d_in[1];          // [2, E]
    const float* ew  = (const float*)d_in[2];
    const float* Wi  = (const float*)d_in[3];
    const float* bi  = (const float*)d_in[4];
    const float* Wh  = (const float*)d_in[5];        // [4, 50, 50]
    const float* bh  = (const float*)d_in[6];        // [4, 50]
    const float* Wf  = (const float*)d_in[7];
    const float* bf  = (const float*)d_in[8];
    const int* row = ei;
    const int* col = ei + E;
    float* out = (float*)d_out;

    // workspace layout
    float* ws   = (float*)d_ws;
    float* dinv = ws;                                  // N
    float* nrm  = dinv + N;                            // E
    float* bufA = nrm + E;                             // N*64 (pre-activation / final agg)
    float* bufB = bufA + (size_t)N * 64;               // N*64 (h / agg)
    float* zf   = bufB + (size_t)N * 64;               // N*16 (final logits pre-agg)

    const int BT = 256;
    dim3 blk(BT);
    auto cdiv = [](long long a, long long b) { return (unsigned)((a + b - 1) / b); };

    // ---- normalization precompute
    k_zero<<<cdiv(N, BT), blk, 0, stream>>>(dinv, (size_t)N);
    k_deg<<<cdiv(E, BT), blk, 0, stream>>>(dinv, col, ew, E);
    k_dinv<<<cdiv(N, BT), blk, 0, stream>>>(dinv, N);
    k_norm<<<cdiv(E, BT), blk, 0, stream>>>(nrm, dinv, row, col, ew, E);

    const unsigned g32  = cdiv(N, 32);       // gemm blocks, 4 col-tiles (32 rows/block)
    const unsigned g128 = cdiv(N, 128);      // gemm blocks, 1 col-tile (128 rows/block)
    const unsigned gS64 = cdiv((long long)E * 32, BT);   // stride 64 -> 32 thr/edge
    const unsigned gS16 = cdiv((long long)E * 8,  BT);   // stride 16 ->  8 thr/edge
    const unsigned gC64 = cdiv((long long)N * 64, BT);
    const unsigned gC16 = cdiv((long long)N * 16, BT);
    const size_t lds784 = (size_t)KCHUNK * 64 * 4;     // 28 KB
    const size_t lds64  = (size_t)64 * 64 * 4;         // 16 KB

    // ---- layer 1: x @ Wi  -> bufA ; aggregate -> bufB ; combine(relu) -> h in bufB
    k_gemm_wmma<<<g32, blk, lds784, stream>>>(x, FEAT, N, Wi, FEAT, HID, FEAT, bufA, 64, 4);
    k_zero<<<cdiv((long long)N * 64, BT), blk, 0, stream>>>(bufB, (size_t)N * 64);
    k_scatter<<<gS64, blk, 0, stream>>>(bufB, bufA, nrm, row, col, E, 6, HID);
    k_combine<<<gC64, blk, 0, stream>>>(bufB, bufA, dinv, bi, N, 6, HID, 1);

    // ---- hidden layers: h(bufB) @ Wh[i] -> bufA ; agg into zeroed bufB ; combine
    for (int i = 0; i < NL; ++i) {
        k_gemm_wmma<<<g32, blk, lds64, stream>>>(bufB, 64, N, Wh + (size_t)i * HID * HID,
                                                 HID, HID, 64, bufA, 64, 4);
        k_zero<<<cdiv((long long)N * 64, BT), blk, 0, stream>>>(bufB, (size_t)N * 64);
        k_scatter<<<gS64, blk, 0, stream>>>(bufB, bufA, nrm, row, col, E, 6, HID);
        k_combine<<<gC64, blk, 0, stream>>>(bufB, bufA, dinv, bh + (size_t)i * HID,
                                            N, 6, HID, 1);
    }

    // ---- final layer: h @ Wf -> zf[N,16] ; agg in bufA[N,16] ; combine (no relu)
    k_gemm_wmma<<<g128, blk, lds64, stream>>>(bufB, 64, N, Wf, HID, CLS, 64, zf, 16, 1);
    k_zero<<<cdiv((long long)N * 16, BT), blk, 0, stream>>>(bufA, (size_t)N * 16);
    k_scatter<<<gS16, blk, 0, stream>>>(bufA, zf, nrm, row, col, E, 4, CLS);
    k_combine<<<gC16, blk, 0, stream>>>(bufA, zf, dinv, bf, N, 4, CLS, 0);

    // ---- log_softmax -> d_out [N,10]
    k_logsoftmax<<<cdiv(N, BT), blk, 0, stream>>>(bufA, out, N);
}